// SemanticBranch_26259430048200
// MI455X (gfx1250) — compile-verified
//
#include <hip/hip_runtime.h>

typedef __attribute__((ext_vector_type(2))) float v2f;
typedef __attribute__((ext_vector_type(8))) float v8f;

#define WPB 8   // waves per block (256 threads)
#define TPW 4   // 16-row tiles per wave

// ---------------- utility kernels ----------------

__global__ void fill_u32(unsigned* __restrict__ p, unsigned v, int n) {
    int i = blockIdx.x * blockDim.x + threadIdx.x;
    if (i < n) p[i] = v;
}

// ---------------- BEV projection (segment max) ----------------

__device__ __forceinline__ unsigned enc_f32(float f) {
    unsigned u = __float_as_uint(f);
    return (u & 0x80000000u) ? ~u : (u | 0x80000000u);
}
__device__ __forceinline__ float dec_f32(unsigned e) {
    return (e & 0x80000000u) ? __uint_as_float(e ^ 0x80000000u)
                             : __uint_as_float(~e);
}

// one thread per (point, channel)
__global__ void bev_scatter(const float* __restrict__ feat,
                            const int* __restrict__ coord,
                            unsigned* __restrict__ bev,
                            int total, int cshift, int cmask, int C, int Y, int X) {
    int t = blockIdx.x * blockDim.x + threadIdx.x;
    if (t >= total) return;
    int p = t >> cshift;
    int c = t & cmask;
    int b = coord[p * 4 + 0];
    int y = coord[p * 4 + 2];
    int x = coord[p * 4 + 3];
    size_t o = (((size_t)b * C + c) * Y + y) * X + x;
    atomicMax(&bev[o], enc_f32(feat[t]));
}

__global__ void bev_finalize(float* __restrict__ bev, int n) {
    int i = blockIdx.x * blockDim.x + threadIdx.x;
    if (i >= n) return;
    unsigned u = ((unsigned*)bev)[i];
    float f = dec_f32(u);
    bool fin = (__float_as_uint(f) & 0x7F800000u) != 0x7F800000u;
    bev[i] = fin ? f : 0.0f;
}

// ---------------- voxel semantic targets ----------------

__global__ void lbl_scatter(const int* __restrict__ pcoord,
                            const int* __restrict__ plabel,
                            unsigned* __restrict__ counts,
                            int np, int Z, int Y, int X) {
    int p = blockIdx.x * blockDim.x + threadIdx.x;
    if (p >= np) return;
    int b = pcoord[p * 4 + 0], z = pcoord[p * 4 + 1];
    int y = pcoord[p * 4 + 2], x = pcoord[p * 4 + 3];
    size_t cell = ((((size_t)b * Z + z) * Y + y) * X + x);
    atomicAdd(&counts[cell * 20 + plabel[p]], 1u);
}

__global__ void lbl_gather(const int* __restrict__ fcoord,
                           const unsigned* __restrict__ counts,
                           float* __restrict__ out,
                           int m, int Z, int Y, int X) {
    int i = blockIdx.x * blockDim.x + threadIdx.x;
    if (i >= m) return;
    int b = fcoord[i * 4 + 0], z = fcoord[i * 4 + 1];
    int y = fcoord[i * 4 + 2], x = fcoord[i * 4 + 3];
    size_t cell = ((((size_t)b * Z + z) * Y + y) * X + x);
    const unsigned* c = counts + cell * 20;
    unsigned best = 0, tot = 0;
    int maj = 0;
#pragma unroll
    for (int l = 0; l < 20; ++l) {
        unsigned v = c[l];
        tot += v;
        if (v > best) { best = v; maj = l; }  // first-max tie-break (jnp.argmax)
    }
    out[i] = (tot > 0) ? (float)maj : 0.0f;
}

// ---------------- heads: x@W1 via fp32 WMMA ----------------

__device__ __forceinline__ v8f wmma_f32(v2f a, v2f b, v8f c) {
    return __builtin_amdgcn_wmma_f32_16x16x4_f32(
        false, a, false, b, (short)0, c, false, false);
}

// pass 1: per-wave partial column sums and sum-of-squares of h = x@W1
template <int CIN>
__global__ __launch_bounds__(256) void head_pass1(const float* __restrict__ x,
                                                  const float* __restrict__ w1,
                                                  float* __restrict__ partials,
                                                  int ntiles) {
    int wave = threadIdx.x >> 5;
    int lane = threadIdx.x & 31;
    int slot = blockIdx.x * WPB + wave;
    int t0 = slot * TPW;
    if (t0 >= ntiles) return;                 // wave-uniform: EXEC stays all-ones
    int tend = min(t0 + TPW, ntiles);
    int mrow = lane & 15;
    int koff = (lane >> 4) << 1;              // 0 for lanes 0-15, 2 for lanes 16-31

    float sum4[4] = {0.f, 0.f, 0.f, 0.f};
    float sq4[4]  = {0.f, 0.f, 0.f, 0.f};

    for (int ti = t0; ti < tend; ++ti) {
        v8f acc[4];
#pragma unroll
        for (int t = 0; t < 4; ++t)
#pragma unroll
            for (int v = 0; v < 8; ++v) acc[t][v] = 0.0f;

        const float* xrow = x + (size_t)(ti * 16 + mrow) * CIN;
#pragma unroll 2
        for (int k0 = 0; k0 < CIN; k0 += 4) {
            v2f a;
            a.x = xrow[k0 + koff];
            a.y = xrow[k0 + koff + 1];
#pragma unroll
            for (int t = 0; t < 4; ++t) {
                int n = (lane & 15) + (t << 4);
                v2f bb;
                bb.x = w1[(k0 + koff) * 64 + n];
                bb.y = w1[(k0 + koff + 1) * 64 + n];
                acc[t] = wmma_f32(a, bb, acc[t]);
            }
        }
#pragma unroll
        for (int t = 0; t < 4; ++t) {
            float s = 0.f, q = 0.f;
#pragma unroll
            for (int v = 0; v < 8; ++v) { float h = acc[t][v]; s += h; q += h * h; }
            sum4[t] += s;
            sq4[t]  += q;
        }
    }
#pragma unroll
    for (int t = 0; t < 4; ++t) {
        float s = sum4[t] + __shfl_xor(sum4[t], 16, 32);
        float q = sq4[t]  + __shfl_xor(sq4[t], 16, 32);
        if (lane < 16) {
            partials[(size_t)slot * 128 + t * 16 + lane]      = s;
            partials[(size_t)slot * 128 + 64 + t * 16 + lane] = q;
        }
    }
}

// tiny deterministic reduction: slots summed in fixed order
__global__ void head_stats(const float* __restrict__ partials, int nslots,
                           float invM, float* __restrict__ mu,
                           float* __restrict__ rstd) {
    int j = threadIdx.x;
    if (j >= 64) return;
    float s = 0.f, q = 0.f;
    for (int n = 0; n < nslots; ++n) {
        s += partials[(size_t)n * 128 + j];
        q += partials[(size_t)n * 128 + 64 + j];
    }
    float m = s * invM;
    float v = q * invM - m * m;
    mu[j]   = m;
    rstd[j] = rsqrtf(v + 1e-5f);
}

// pass 2: recompute h via WMMA, normalize + leaky ReLU, then @W2 + b2
template <int CIN>
__global__ __launch_bounds__(256) void head_pass2(const float* __restrict__ x,
                                                  const float* __restrict__ w1,
                                                  const float* __restrict__ mu,
                                                  const float* __restrict__ rstd,
                                                  const float* __restrict__ gg,
                                                  const float* __restrict__ bb,
                                                  const float* __restrict__ w2,
                                                  const float* __restrict__ b2,
                                                  float* __restrict__ out,
                                                  int ntiles) {
    __shared__ float sh[WPB][16][68];
    int wave = threadIdx.x >> 5;
    int lane = threadIdx.x & 31;
    int slot = blockIdx.x * WPB + wave;
    int t0 = slot * TPW;
    if (t0 >= ntiles) return;                 // wave-uniform
    int tend = min(t0 + TPW, ntiles);
    int mrow = lane & 15;
    int koff = (lane >> 4) << 1;
    int rbase = (lane >> 4) << 3;             // 0 or 8 (C/D layout: lanes 16-31 = M+8)

    float scale[4], shift[4];
#pragma unroll
    for (int t = 0; t < 4; ++t) {
        int j = t * 16 + (lane & 15);
        float sc = rstd[j] * gg[j];
        scale[t] = sc;
        shift[t] = bb[j] - mu[j] * sc;
    }

    for (int ti = t0; ti < tend; ++ti) {
        v8f acc[4];
#pragma unroll
        for (int t = 0; t < 4; ++t)
#pragma unroll
            for (int v = 0; v < 8; ++v) acc[t][v] = 0.0f;

        const float* xrow = x + (size_t)(ti * 16 + mrow) * CIN;
#pragma unroll 2
        for (int k0 = 0; k0 < CIN; k0 += 4) {
            v2f a;
            a.x = xrow[k0 + koff];
            a.y = xrow[k0 + koff + 1];
#pragma unroll
            for (int t = 0; t < 4; ++t) {
                int n = (lane & 15) + (t << 4);
                v2f bv;
                bv.x = w1[(k0 + koff) * 64 + n];
                bv.y = w1[(k0 + koff + 1) * 64 + n];
                acc[t] = wmma_f32(a, bv, acc[t]);
            }
        }
        // normalize + leaky relu -> LDS tile (wave-private; LDS ops in-order per wave)
#pragma unroll
        for (int t = 0; t < 4; ++t) {
            int j = t * 16 + (lane & 15);
#pragma unroll
            for (int v = 0; v < 8; ++v) {
                float h = acc[t][v] * scale[t] + shift[t];
                h = (h >= 0.f) ? h : 0.1f * h;
                sh[wave][v + rbase][j] = h;
            }
        }
        // 16 rows x 19 outputs, dot over 64
        for (int idx = lane; idx < 16 * 19; idx += 32) {
            int r = idx / 19;
            int c = idx - r * 19;
            float o = b2[c];
#pragma unroll 8
            for (int k = 0; k < 64; ++k) o += sh[wave][r][k] * w2[k * 19 + c];
            out[(size_t)(ti * 16 + r) * 19 + c] = o;
        }
    }
}

// ---------------- launch ----------------

static inline int cdiv(long a, long b) { return (int)((a + b - 1) / b); }

extern "C" void kernel_launch(void* const* d_in, const int* in_sizes, int n_in,
                              void* d_out, int out_size, void* d_ws, size_t ws_size,
                              hipStream_t stream) {
    (void)in_sizes; (void)n_in; (void)out_size; (void)ws_size;
    const float* proj1 = (const float*)d_in[0];
    const float* proj2 = (const float*)d_in[1];
    const float* proj3 = (const float*)d_in[2];
    const int* vw1c = (const int*)d_in[3];
    const int* vw2c = (const int*)d_in[4];
    const int* vw3c = (const int*)d_in[5];
    const int* idx02 = (const int*)d_in[6];
    const int* idx04 = (const int*)d_in[7];
    const int* idx08 = (const int*)d_in[8];
    const int* plab = (const int*)d_in[9];
    const float* h2w1 = (const float*)d_in[10];
    const float* h2g  = (const float*)d_in[11];
    const float* h2b  = (const float*)d_in[12];
    const float* h2w2 = (const float*)d_in[13];
    const float* h2b2 = (const float*)d_in[14];
    const float* h4w1 = (const float*)d_in[15];
    const float* h4g  = (const float*)d_in[16];
    const float* h4b  = (const float*)d_in[17];
    const float* h4w2 = (const float*)d_in[18];
    const float* h4b2 = (const float*)d_in[19];
    const float* h8w1 = (const float*)d_in[20];
    const float* h8g  = (const float*)d_in[21];
    const float* h8b  = (const float*)d_in[22];
    const float* h8w2 = (const float*)d_in[23];
    const float* h8b2 = (const float*)d_in[24];

    float* out = (float*)d_out;
    const int M1 = 200000, M2 = 100000, M3 = 50000, NP = 500000;
    // output offsets
    const size_t BEV1 = 0, BEV2 = 2097152, BEV3 = 3145728;
    const size_t BEVTOT = 3670016;
    const size_t LOG2 = 3670016, LOG4 = 7470016, LOG8 = 9370016;
    const size_t LBL2 = 10320016, LBL4 = 10520016, LBL8 = 10620016;

    // workspace layout
    unsigned* counts = (unsigned*)d_ws;                       // up to 10,485,760 u32
    float* stats = (float*)((char*)d_ws + 10485760ull * 4);   // 3 * 128 floats
    const int NS2 = cdiv(M1 / 16, TPW), NS4 = cdiv(M2 / 16, TPW), NS8 = cdiv(M3 / 16, TPW);
    float* part2 = stats + 3 * 128;
    float* part4 = part2 + (size_t)NS2 * 128;
    float* part8 = part4 + (size_t)NS4 * 128;

    const int T = 256;

    // ---- BEV: init (encoded -inf), scatter, finalize ----
    fill_u32<<<cdiv(BEVTOT, T), T, 0, stream>>>((unsigned*)out, 0x007FFFFFu, (int)BEVTOT);
    bev_scatter<<<cdiv((long)M1 * 64, T), T, 0, stream>>>(proj1, vw1c, (unsigned*)(out + BEV1),
                                                          M1 * 64, 6, 63, 64, 128, 128);
    bev_scatter<<<cdiv((long)M2 * 128, T), T, 0, stream>>>(proj2, vw2c, (unsigned*)(out + BEV2),
                                                           M2 * 128, 7, 127, 128, 64, 64);
    bev_scatter<<<cdiv((long)M3 * 256, T), T, 0, stream>>>(proj3, vw3c, (unsigned*)(out + BEV3),
                                                           M3 * 256, 8, 255, 256, 32, 32);
    bev_finalize<<<cdiv(BEVTOT, T), T, 0, stream>>>(out, (int)BEVTOT);

    // ---- labels (counts buffer reused sequentially) ----
    {   // level 2: Z=16,Y=128,X=128
        int nc = 2 * 16 * 128 * 128 * 20;
        fill_u32<<<cdiv(nc, T), T, 0, stream>>>(counts, 0u, nc);
        lbl_scatter<<<cdiv(NP, T), T, 0, stream>>>(idx02, plab, counts, NP, 16, 128, 128);
        lbl_gather<<<cdiv(M1, T), T, 0, stream>>>(vw1c, counts, out + LBL2, M1, 16, 128, 128);
    }
    {   // level 4: Z=8,Y=64,X=64
        int nc = 2 * 8 * 64 * 64 * 20;
        fill_u32<<<cdiv(nc, T), T, 0, stream>>>(counts, 0u, nc);
        lbl_scatter<<<cdiv(NP, T), T, 0, stream>>>(idx04, plab, counts, NP, 8, 64, 64);
        lbl_gather<<<cdiv(M2, T), T, 0, stream>>>(vw2c, counts, out + LBL4, M2, 8, 64, 64);
    }
    {   // level 8: Z=4,Y=32,X=32
        int nc = 2 * 4 * 32 * 32 * 20;
        fill_u32<<<cdiv(nc, T), T, 0, stream>>>(counts, 0u, nc);
        lbl_scatter<<<cdiv(NP, T), T, 0, stream>>>(idx08, plab, counts, NP, 4, 32, 32);
        lbl_gather<<<cdiv(M3, T), T, 0, stream>>>(vw3c, counts, out + LBL8, M3, 4, 32, 32);
    }

    // ---- heads ----
    {   // head2: Cin=64
        int ntiles = M1 / 16, blocks = cdiv(NS2, WPB);
        head_pass1<64><<<blocks, T, 0, stream>>>(proj1, h2w1, part2, ntiles);
        head_stats<<<1, 64, 0, stream>>>(part2, NS2, 1.0f / M1, stats + 0, stats + 64);
        head_pass2<64><<<blocks, T, 0, stream>>>(proj1, h2w1, stats + 0, stats + 64,
                                                 h2g, h2b, h2w2, h2b2, out + LOG2, ntiles);
    }
    {   // head4: Cin=128
        int ntiles = M2 / 16, blocks = cdiv(NS4, WPB);
        head_pass1<128><<<blocks, T, 0, stream>>>(proj2, h4w1, part4, ntiles);
        head_stats<<<1, 64, 0, stream>>>(part4, NS4, 1.0f / M2, stats + 128, stats + 192);
        head_pass2<128><<<blocks, T, 0, stream>>>(proj2, h4w1, stats + 128, stats + 192,
                                                  h4g, h4b, h4w2, h4b2, out + LOG4, ntiles);
    }
    {   // head8: Cin=256
        int ntiles = M3 / 16, blocks = cdiv(NS8, WPB);
        head_pass1<256><<<blocks, T, 0, stream>>>(proj3, h8w1, part8, ntiles);
        head_stats<<<1, 64, 0, stream>>>(part8, NS8, 1.0f / M3, stats + 256, stats + 320);
        head_pass2<256><<<blocks, T, 0, stream>>>(proj3, h8w1, stats + 256, stats + 320,
                                                  h8g, h8b, h8w2, h8b2, out + LOG8, ntiles);
    }
}